// BatchedDynamicBicycleModel_88021059764785
// MI455X (gfx1250) — compile-verified
//
#include <hip/hip_runtime.h>
#include <hip/hip_bf16.h>
#include <stdint.h>

#define AS1 __attribute__((address_space(1)))
#define AS3 __attribute__((address_space(3)))

typedef int v4i __attribute__((vector_size(16)));   // matches async builtin param type

// ---- CDNA5 async LDS datapath (guarded; falls back to plain copies) ----
#ifndef __HIP_DEVICE_COMPILE__
  #define USE_ASYNC_LDS 0
  #define WAIT_ASYNC(n) do {} while (0)
#else
  #if __has_builtin(__builtin_amdgcn_global_load_async_to_lds_b128) && \
      __has_builtin(__builtin_amdgcn_global_store_async_from_lds_b128)
    #define USE_ASYNC_LDS 1
  #else
    #define USE_ASYNC_LDS 0
  #endif
  #if __has_builtin(__builtin_amdgcn_s_wait_asynccnt)
    #define WAIT_ASYNC(n) __builtin_amdgcn_s_wait_asynccnt(n)
  #else
    #define WAIT_ASYNC(n) asm volatile("s_wait_asynccnt %0" ::"i"(n) : "memory")
  #endif
#endif

#define MAX_STEER_F 0.52359879f    // 30*pi/180 in fp32
#define FYF_MAX_F   7567.71428571f
#define FYR_MAX_F   5675.78571429f
#define VEFF_MIN_F  5.5555553f     // 20/3.6
#define H_STEP      0.01f          // all 10 substeps use h = fp32(0.01)

__device__ __forceinline__ float clampf(float x, float lo, float hi) {
  return fminf(fmaxf(x, lo), hi);
}

__device__ __forceinline__ float fast_rcp(float x) {
#if defined(__HIP_DEVICE_COMPILE__) && __has_builtin(__builtin_amdgcn_rcpf)
  return __builtin_amdgcn_rcpf(x);   // v_rcp_f32; v_eff >= 5.55 so well conditioned
#else
  return 1.0f / x;
#endif
}

// 10 unrolled substeps on registers.
__device__ __forceinline__ void integrate10(float& x, float& y, float& psi, float& v,
                                            float& a, float& delta, float& beta, float& r,
                                            float a_ref, float dref) {
  v     = fmaxf(v, 0.0f);                                   // reset() postprocess
  delta = clampf(delta, -MAX_STEER_F, MAX_STEER_F);
  const float ar01 = a_ref * 0.1f;                          // hoisted lag targets
  const float dr01 = dref  * 0.1f;
#pragma unroll
  for (int k = 0; k < 10; ++k) {
    const float v_eff   = fmaxf(v, VEFF_MIN_F);
    const float inv_v   = fast_rcp(v_eff);
    const float rv      = r * inv_v;
    const float alpha_f = fmaf(1.2f, rv, beta) - delta;
    const float alpha_r = fmaf(-1.6f, rv, beta);
    const float F_yf    = clampf(-80000.0f * alpha_f, -FYF_MAX_F, FYF_MAX_F);
    const float F_yr    = clampf(-80000.0f * alpha_r, -FYR_MAX_F, FYR_MAX_F);
    const float Fsum    = F_yf + F_yr;
    const float psi_eff = psi + beta;
    const float sp = __sinf(psi_eff);                        // v_sin_f32
    const float cp = __cosf(psi_eff);                        // v_cos_f32 (shared *1/2pi CSEs)
    const float vh = v * H_STEP;
    x    = fmaf(vh, cp, x);                                  // old v, old psi/beta
    y    = fmaf(vh, sp, y);
    psi  = fmaf(r, H_STEP, psi);                             // old r
    beta = fmaf(Fsum * inv_v, (H_STEP / 1500.0f),            // h/M folded
                fmaf(-H_STEP, r, beta));                     // old r
    r    = fmaf(fmaf(1.2f, F_yf, -1.6f * F_yr),
                (H_STEP / 2250.0f), r);                      // h/IZ folded
    v    = fmaxf(fmaf(a, H_STEP, v), 0.0f);                  // old a
    a    = fmaf(a, 0.9f, ar01);                              // first-order lag
    delta = clampf(fmaf(delta, 0.9f, dr01), -MAX_STEER_F, MAX_STEER_F);
  }
}

// Fast path: B % 256 == 0. Block = 256 threads = 256 rows = 2304 dwords = 576 x 16B chunks.
// Staging: thread t owns chunks t, t+256, and (t<64) t+512; the async instruction's
// IOFFSET is added to BOTH the LDS and global addresses (ISA 15.18.3), and our LDS and
// global offsets advance in lockstep, so one VGPR address pair + imm 0/4096/8192 covers it.
__launch_bounds__(256)
__global__ void bike10_fast(const float* __restrict__ st,
                            const float* __restrict__ act,
                            float* __restrict__ out) {
  __shared__ float lds[2304];
  const int t      = threadIdx.x;
  const int row    = (blockIdx.x << 8) + t;
  const int dwBase = (blockIdx.x << 8) * 9;

#if USE_ASYNC_LDS
  {
    AS1 v4i* gin = (AS1 v4i*)(uintptr_t)(st + dwBase) + t;   // 16B-aligned (9216 | block)
    AS3 v4i* lin = (AS3 v4i*)(uintptr_t)(&lds[t << 2]);
    __builtin_amdgcn_global_load_async_to_lds_b128(gin, lin, 0, 0);
    __builtin_amdgcn_global_load_async_to_lds_b128(gin, lin, 4096, 0);
    if (t < 64)   // waves 0-1 only, non-divergent
      __builtin_amdgcn_global_load_async_to_lds_b128(gin, lin, 8192, 0);
  }
  WAIT_ASYNC(0);
#else
  {
    const float4* gin = (const float4*)(st + dwBase) + t;
    float4*       lin = (float4*)(&lds[t << 2]);
    lin[0]   = gin[0];
    lin[256] = gin[256];
    if (t < 64) lin[512] = gin[512];
  }
#endif
  __syncthreads();

  const float* s = &lds[t * 9];     // stride-9 dwords: conflict-free on 64 banks
  float x = s[0], y = s[1], psi = s[2], v = s[3];
  float a = s[4], delta = s[5], beta = s[6], r = s[7];
  // s[8] (delta_ref) is dead: overwritten by the clipped action.

  const float2 av = *(const float2*)(act + row * 2);
  const float a_ref = clampf(av.x, -6.0f, 3.0f);
  const float dref  = clampf(av.y, -MAX_STEER_F, MAX_STEER_F);

  integrate10(x, y, psi, v, a, delta, beta, r, a_ref, dref);

  float* o = &lds[t * 9];           // private region: only this thread read it
  o[0] = x; o[1] = y; o[2] = psi; o[3] = v;
  o[4] = a; o[5] = delta; o[6] = beta; o[7] = r; o[8] = dref;
  __syncthreads();

#if USE_ASYNC_LDS
  {
    AS1 v4i* gout = (AS1 v4i*)(uintptr_t)(out + dwBase) + t;
    AS3 v4i* lout = (AS3 v4i*)(uintptr_t)(&lds[t << 2]);
    __builtin_amdgcn_global_store_async_from_lds_b128(gout, lout, 0, 0);
    __builtin_amdgcn_global_store_async_from_lds_b128(gout, lout, 4096, 0);
    if (t < 64)
      __builtin_amdgcn_global_store_async_from_lds_b128(gout, lout, 8192, 0);
  }
  WAIT_ASYNC(0);
#else
  {
    float4*       gout = (float4*)(out + dwBase) + t;
    const float4* lout = (const float4*)(&lds[t << 2]);
    gout[0]   = lout[0];
    gout[256] = lout[256];
    if (t < 64) gout[512] = lout[512];
  }
#endif
}

// Generic tail-safe kernel (only used if B % 256 != 0; never for this workload).
__launch_bounds__(256)
__global__ void bike10_generic(const float* __restrict__ st,
                               const float* __restrict__ act,
                               float* __restrict__ out, int B) {
  const int row = blockIdx.x * 256 + threadIdx.x;
  if (row >= B) return;
  const float* s = st + row * 9;
  float x = s[0], y = s[1], psi = s[2], v = s[3];
  float a = s[4], delta = s[5], beta = s[6], r = s[7];
  const float2 av = *(const float2*)(act + row * 2);
  const float a_ref = clampf(av.x, -6.0f, 3.0f);
  const float dref  = clampf(av.y, -MAX_STEER_F, MAX_STEER_F);
  integrate10(x, y, psi, v, a, delta, beta, r, a_ref, dref);
  float* o = out + row * 9;
  o[0] = x; o[1] = y; o[2] = psi; o[3] = v;
  o[4] = a; o[5] = delta; o[6] = beta; o[7] = r; o[8] = dref;
}

extern "C" void kernel_launch(void* const* d_in, const int* in_sizes, int n_in,
                              void* d_out, int out_size, void* d_ws, size_t ws_size,
                              hipStream_t stream) {
  const float* init_state = (const float*)d_in[0];  // (B, 9) fp32
  const float* action     = (const float*)d_in[1];  // (B, 2) fp32
  float* out = (float*)d_out;                       // (B, 9) fp32

  const int B = in_sizes[0] / 9;                    // 4,194,304
  if ((B & 255) == 0) {
    bike10_fast<<<B >> 8, 256, 0, stream>>>(init_state, action, out);
  } else {
    bike10_generic<<<(B + 255) / 256, 256, 0, stream>>>(init_state, action, out, B);
  }
}